// RNAGNN_3453153706245
// MI455X (gfx1250) — compile-verified
//
#include <hip/hip_runtime.h>
#include <hip/hip_bf16.h>
#include <math.h>

typedef float v2f __attribute__((ext_vector_type(2)));
typedef float v8f __attribute__((ext_vector_type(8)));

#define NEG_SLOPE 0.2f

__device__ __forceinline__ float leaky(float x) { return x > 0.f ? x : NEG_SLOPE * x; }

// float atomic-max via monotone int mapping (works for mixed signs)
__device__ __forceinline__ void atomicFMax(float* addr, float v) {
    if (v >= 0.f) atomicMax((int*)addr, __float_as_int(v));
    else          atomicMin((unsigned int*)addr, __float_as_uint(v));
}

// K0: tbl[4][16] = nt_emb[4][32] @ gcn_w[32][16]  (embedding lookup folds GCN GEMM)
__global__ void k_tbl(const float* __restrict__ nt_emb, const float* __restrict__ gcn_w,
                      float* __restrict__ tbl) {
    int t = threadIdx.x >> 4, d = threadIdx.x & 15;
    if (threadIdx.x < 64) {
        float s = 0.f;
        for (int k = 0; k < 32; ++k) s += nt_emb[t * 32 + k] * gcn_w[k * 16 + d];
        tbl[t * 16 + d] = s;
    }
}

// K1: deg init (self loop => 1)
__global__ void k_deg_init(float* __restrict__ deg, int n) {
    int i = blockIdx.x * blockDim.x + threadIdx.x;
    if (i < n) deg[i] = 1.0f;
}

// K2: degree count over edges
__global__ void k_deg_edge(const int* __restrict__ ei, float* __restrict__ deg, int E) {
    int e = blockIdx.x * blockDim.x + threadIdx.x;
    if (e < E) atomicAdd(&deg[ei[E + e]], 1.0f);
}

// K3: dinv = rsqrt(deg); hn[i] = tbl[x_nt[i]] * dinv[i]; hsum init = self-loop term tbl*dinv^2
__global__ void k_node_h(const int* __restrict__ x_nt, const float* __restrict__ tbl,
                         float* __restrict__ dinv, float* __restrict__ hn,
                         float* __restrict__ hsum, int n) {
    int i = blockIdx.x * blockDim.x + threadIdx.x;
    if (i >= n) return;
    float di = rsqrtf(dinv[i]);   // dinv buffer holds deg on entry
    dinv[i] = di;
    const float* tr = tbl + x_nt[i] * 16;
    for (int d = 0; d < 16; ++d) {
        float v = tr[d] * di;
        hn[(size_t)i * 16 + d]   = v;
        hsum[(size_t)i * 16 + d] = v * di;
    }
}

// K4: GCN scatter: hsum[dst] += hn[src] * dinv[dst]
__global__ void k_gcn_edge(const int* __restrict__ ei, const float* __restrict__ dinv,
                           const float* __restrict__ hn, float* __restrict__ hsum, int E) {
    int e = blockIdx.x * blockDim.x + threadIdx.x;
    if (e >= E) return;
    int s = ei[e], d = ei[E + e];
    float w = dinv[d];
    const float4* hr = (const float4*)(hn + (size_t)s * 16);
    float4 h0 = hr[0], h1 = hr[1], h2 = hr[2], h3 = hr[3];
    float v[16] = { h0.x,h0.y,h0.z,h0.w, h1.x,h1.y,h1.z,h1.w,
                    h2.x,h2.y,h2.z,h2.w, h3.x,h3.y,h3.z,h3.w };
    float* o = hsum + (size_t)d * 16;
    for (int k = 0; k < 16; ++k) atomicAdd(&o[k], v[k] * w);
}

// K5: per 16-node tile, fp32 WMMA: G = (hsum + gcn_b) @ gat_w; attention dots; init m = self-edge logit
__global__ __launch_bounds__(32) void k_gat(const float* __restrict__ hsum,
                                            const float* __restrict__ gcn_b,
                                            const float* __restrict__ gat_w,
                                            const float* __restrict__ att_src,
                                            const float* __restrict__ att_dst,
                                            float* __restrict__ g, float* __restrict__ as_,
                                            float* __restrict__ ad_, float* __restrict__ mx,
                                            int n) {
    int lane = threadIdx.x;
    int base = blockIdx.x * 16;
    int mrow = lane & 15;
    int koff = (lane < 16) ? 0 : 2;
    int hi8  = (lane < 16) ? 0 : 8;
    int node = base + mrow; if (node >= n) node = n - 1;   // clamp, keep EXEC full

    v8f c = {};
    for (int ks = 0; ks < 4; ++ks) {
        int kk = ks * 4 + koff;
        v2f a, b;
        a.x = hsum[(size_t)node * 16 + kk]     + gcn_b[kk];
        a.y = hsum[(size_t)node * 16 + kk + 1] + gcn_b[kk + 1];
        b.x = gat_w[kk * 16 + mrow];
        b.y = gat_w[(kk + 1) * 16 + mrow];
        c = __builtin_amdgcn_wmma_f32_16x16x4_f32(false, a, false, b, (short)0, c, false, false);
    }
    // attention dot-products via cross-lane reduction over the 16-lane N groups
    float es = att_src[mrow], ed = att_dst[mrow];
    float asv[8], adv[8];
    for (int r = 0; r < 8; ++r) { asv[r] = c[r] * es; adv[r] = c[r] * ed; }
    for (int m = 1; m <= 8; m <<= 1)
        for (int r = 0; r < 8; ++r) {
            asv[r] += __shfl_xor(asv[r], m, 32);
            adv[r] += __shfl_xor(adv[r], m, 32);
        }
    for (int r = 0; r < 8; ++r) {
        int row = base + r + hi8;
        if (row < n) g[(size_t)row * 16 + mrow] = c[r];
    }
    if (mrow == 0) {
        for (int r = 0; r < 8; ++r) {
            int row = base + r + hi8;
            if (row < n) {
                as_[row] = asv[r];
                ad_[row] = adv[r];
                mx[row]  = leaky(asv[r] + adv[r]);   // self-loop logit seeds segment max
            }
        }
    }
}

// K6: segment max over edges
__global__ void k_att_edge(const int* __restrict__ ei, const float* __restrict__ as_,
                           const float* __restrict__ ad_, float* __restrict__ mx, int E) {
    int e = blockIdx.x * blockDim.x + threadIdx.x;
    if (e >= E) return;
    int s = ei[e], d = ei[E + e];
    atomicFMax(&mx[d], leaky(as_[s] + ad_[d]));
}

// K7: self-loop softmax term: s = exp(e_self - m); anum = that * g[i]
__global__ void k_self(const float* __restrict__ as_, const float* __restrict__ ad_,
                       const float* __restrict__ mx, const float* __restrict__ g,
                       float* __restrict__ ss, float* __restrict__ anum, int n) {
    int i = blockIdx.x * blockDim.x + threadIdx.x;
    if (i >= n) return;
    float ex = __expf(leaky(as_[i] + ad_[i]) - mx[i]);
    ss[i] = ex;
    for (int d = 0; d < 16; ++d)
        anum[(size_t)i * 16 + d] = ex * g[(size_t)i * 16 + d];
}

// K8: edge softmax accumulation: ss[dst] += ex; anum[dst] += ex * g[src]
__global__ void k_gat_edge(const int* __restrict__ ei, const float* __restrict__ as_,
                           const float* __restrict__ ad_, const float* __restrict__ mx,
                           const float* __restrict__ g, float* __restrict__ ss,
                           float* __restrict__ anum, int E) {
    int e = blockIdx.x * blockDim.x + threadIdx.x;
    if (e >= E) return;
    int s = ei[e], d = ei[E + e];
    float ex = __expf(leaky(as_[s] + ad_[d]) - mx[d]);
    atomicAdd(&ss[d], ex);
    const float4* gr = (const float4*)(g + (size_t)s * 16);
    float4 g0 = gr[0], g1 = gr[1], g2 = gr[2], g3 = gr[3];
    float v[16] = { g0.x,g0.y,g0.z,g0.w, g1.x,g1.y,g1.z,g1.w,
                    g2.x,g2.y,g2.z,g2.w, g3.x,g3.y,g3.z,g3.w };
    float* o = anum + (size_t)d * 16;
    for (int k = 0; k < 16; ++k) atomicAdd(&o[k], ex * v[k]);
}

// K9: head — alpha normalize + gat_b + relu + conv1d (per node), then fp32 WMMA MLP
//      X[16x16] @ W1^T -> relu -> LDS -> Z[16x64] @ W2^T -> out. 80 WMMAs/tile.
__global__ __launch_bounds__(32) void k_head(const float* __restrict__ anum,
                                             const float* __restrict__ ss,
                                             const float* __restrict__ gat_b,
                                             const float* __restrict__ conv_w,
                                             const float* __restrict__ conv_b,
                                             const float* __restrict__ w1,
                                             const float* __restrict__ b1,
                                             const float* __restrict__ w2,
                                             const float* __restrict__ b2,
                                             float* __restrict__ out, int n) {
    __shared__ float Xs[16][20];   // pad 20: row stride 20 -> conflict-free column reads
    __shared__ float Zs[16][68];   // pad 68: row stride 68 -> conflict-free column reads
    int lane = threadIdx.x;
    int base = blockIdx.x * 16;

    if (lane < 16) {
        int node = base + lane; if (node >= n) node = n - 1;
        float inv = 1.0f / ss[node];
        float x[16];
        for (int d = 0; d < 16; ++d) {
            float v = anum[(size_t)node * 16 + d] * inv + gat_b[d];
            x[d] = v > 0.f ? v : 0.f;
        }
        // conv1d: 8 ch, width 2, kernel 3, pad 1 (OIW weights)
        for (int o = 0; o < 8; ++o) {
            float y0 = conv_b[o], y1 = conv_b[o];
            for (int i = 0; i < 8; ++i) {
                float c0 = conv_w[(o * 8 + i) * 3 + 0];
                float c1 = conv_w[(o * 8 + i) * 3 + 1];
                float c2 = conv_w[(o * 8 + i) * 3 + 2];
                float x0 = x[i * 2 + 0], x1 = x[i * 2 + 1];
                y0 += c1 * x0 + c2 * x1;
                y1 += c0 * x0 + c1 * x1;
            }
            Xs[lane][o * 2 + 0] = y0 > 0.f ? y0 : 0.f;
            Xs[lane][o * 2 + 1] = y1 > 0.f ? y1 : 0.f;
        }
    }
    __syncthreads();

    int mrow = lane & 15;
    int koff = (lane < 16) ? 0 : 2;
    int hi8  = (lane < 16) ? 0 : 8;

    // ---- GEMM1: [16x16] @ W1^T[16x64] -> [16x64], K=16 => 4 k-slices x 4 n-chunks
    v8f acc1[4] = {};
    for (int ks = 0; ks < 4; ++ks) {
        int kk = ks * 4 + koff;
        v2f a; a.x = Xs[mrow][kk]; a.y = Xs[mrow][kk + 1];
        for (int cc = 0; cc < 4; ++cc) {
            int nn = cc * 16 + mrow;
            v2f b; b.x = w1[nn * 16 + kk]; b.y = w1[nn * 16 + kk + 1];
            acc1[cc] = __builtin_amdgcn_wmma_f32_16x16x4_f32(false, a, false, b, (short)0,
                                                             acc1[cc], false, false);
        }
    }
    for (int cc = 0; cc < 4; ++cc) {
        int nn = cc * 16 + mrow;
        float bb = b1[nn];
        for (int r = 0; r < 8; ++r) {
            float v = acc1[cc][r] + bb;
            Zs[r + hi8][nn] = v > 0.f ? v : 0.f;   // D layout -> row-major LDS (A for GEMM2)
        }
    }
    __syncthreads();

    // ---- GEMM2: [16x64] @ W2^T[64x64] -> [16x64], K=64 => 16 k-slices x 4 n-chunks
    v8f acc2[4] = {};
    for (int ks = 0; ks < 16; ++ks) {
        int kk = ks * 4 + koff;
        v2f a; a.x = Zs[mrow][kk]; a.y = Zs[mrow][kk + 1];
        for (int cc = 0; cc < 4; ++cc) {
            int nn = cc * 16 + mrow;
            v2f b; b.x = w2[nn * 64 + kk]; b.y = w2[nn * 64 + kk + 1];
            acc2[cc] = __builtin_amdgcn_wmma_f32_16x16x4_f32(false, a, false, b, (short)0,
                                                             acc2[cc], false, false);
        }
    }
    for (int cc = 0; cc < 4; ++cc) {
        int nn = cc * 16 + mrow;
        float bb = b2[nn];
        for (int r = 0; r < 8; ++r) {
            int row = base + r + hi8;
            if (row < n) out[(size_t)row * 64 + nn] = acc2[cc][r] + bb;
        }
    }
}

extern "C" void kernel_launch(void* const* d_in, const int* in_sizes, int n_in,
                              void* d_out, int out_size, void* d_ws, size_t ws_size,
                              hipStream_t stream) {
    const int*   x_nt    = (const int*)d_in[0];
    const int*   ei      = (const int*)d_in[1];
    const float* nt_emb  = (const float*)d_in[2];
    const float* gcn_w   = (const float*)d_in[3];
    const float* gcn_b   = (const float*)d_in[4];
    const float* gat_w   = (const float*)d_in[5];
    const float* att_src = (const float*)d_in[6];
    const float* att_dst = (const float*)d_in[7];
    const float* gat_b   = (const float*)d_in[8];
    const float* conv_w  = (const float*)d_in[9];
    const float* conv_b  = (const float*)d_in[10];
    const float* w1      = (const float*)d_in[11];
    const float* b1      = (const float*)d_in[12];
    const float* w2      = (const float*)d_in[13];
    const float* b2      = (const float*)d_in[14];
    float* out = (float*)d_out;

    int n = in_sizes[0];
    int E = in_sizes[1] / 2;

    // workspace layout (floats): 64 + 5N + 4*16N  (~27.6 MB for N=100K)
    float* ws   = (float*)d_ws;
    float* tbl  = ws;
    float* dinv = tbl + 64;                 // deg then dinv
    float* as_  = dinv + n;
    float* ad_  = as_ + n;
    float* mx   = ad_ + n;
    float* ss   = mx + n;
    float* hn   = ss + n;                   // N x 16
    float* hsum = hn + (size_t)n * 16;      // N x 16  (GCN result pre-bias)
    float* gfe  = hsum + (size_t)n * 16;    // N x 16  (GAT projected features)
    float* anum = gfe + (size_t)n * 16;     // N x 16  (softmax numerator accum)

    int nb = (n + 255) / 256;
    int eb = (E + 255) / 256;
    int tiles = (n + 15) / 16;

    k_tbl<<<1, 64, 0, stream>>>(nt_emb, gcn_w, tbl);
    k_deg_init<<<nb, 256, 0, stream>>>(dinv, n);
    k_deg_edge<<<eb, 256, 0, stream>>>(ei, dinv, E);
    k_node_h<<<nb, 256, 0, stream>>>(x_nt, tbl, dinv, hn, hsum, n);
    k_gcn_edge<<<eb, 256, 0, stream>>>(ei, dinv, hn, hsum, E);
    k_gat<<<tiles, 32, 0, stream>>>(hsum, gcn_b, gat_w, att_src, att_dst, gfe, as_, ad_, mx, n);
    k_att_edge<<<eb, 256, 0, stream>>>(ei, as_, ad_, mx, E);
    k_self<<<nb, 256, 0, stream>>>(as_, ad_, mx, gfe, ss, anum, n);
    k_gat_edge<<<eb, 256, 0, stream>>>(ei, as_, ad_, mx, gfe, ss, anum, E);
    k_head<<<tiles, 32, 0, stream>>>(anum, ss, gat_b, conv_w, conv_b, w1, b1, w2, b2, out, n);
}